// MWTLayer1d_9663676416254
// MI455X (gfx1250) — compile-verified
//
#include <hip/hip_runtime.h>
#include <math.h>

#define P_  8
#define B_  8
#define C_  128
#define L_  8192
#define LS_ 4096          // scaling-coeff length per row
#define PC_ 1024          // P*C channels

// ---- GEMM tiling ----
#define BM 128
#define BN 128
#define BK 16
#define LDA 20            // padded LDS leading dim for A   (conflict-free, keeps 8B/16B align)
#define LDB 20            // padded LDS leading dim for B^T (column-major B tile)

typedef __attribute__((ext_vector_type(2))) float v2f;
typedef __attribute__((ext_vector_type(4))) float v4f;
typedef __attribute__((ext_vector_type(8))) float v8f;

// -------------------------------------------------------------------------
// Kernel 1: Haar analysis. Each thread consumes one v4f of xs (two k=2
// blocks), writes v2f of sc in coupled-GEMM layout [b][p*C+c][n], and
// directly applies the synthesis matrix S to the adjacent detail pair to
// produce the SECOND half of the output. xs is streamed non-temporally
// (read once); out is stored non-temporally (written once, never re-read);
// sc stays regular-temporal so the GEMM can find it in L2 (192 MB).
// -------------------------------------------------------------------------
__global__ __launch_bounds__(256)
void analysis_kernel(const v4f* __restrict__ xs4,
                     const float* __restrict__ A,
                     const float* __restrict__ S,
                     v2f* __restrict__ sc2,
                     v2f* __restrict__ out2) {
    const long t = (long)blockIdx.x * blockDim.x + threadIdx.x; // P*B*C*2048
    const int  m   = (int)(t & 2047);
    const long idx = t >> 11;                 // (p*B + b)*C + c
    const int  c   = (int)(idx & (C_ - 1));
    const int  pb  = (int)(idx >> 7);
    const int  b   = pb & (B_ - 1);
    const int  p   = pb >> 3;

    const float a00 = A[0], a01 = A[1], a10 = A[2], a11 = A[3];
    const float s00 = S[0], s01 = S[1], s10 = S[2], s11 = S[3];

    const v4f x = __builtin_nontemporal_load(&xs4[t]);   // x[4m..4m+3]
    // analysis
    const float sc0 = a00 * x.x + a01 * x.y;
    const float sc1 = a00 * x.z + a01 * x.w;
    const float d0  = a10 * x.x + a11 * x.y;
    const float d1  = a10 * x.z + a11 * x.w;

    // sc in [b][p*C+c][2m..2m+1] layout for the coupling GEMM
    v2f sv; sv.x = sc0; sv.y = sc1;
    sc2[((size_t)b * PC_ + (size_t)p * C_ + c) * (LS_ / 2) + m] = sv;

    // synthesis on detail pair -> out[..., Ls + 2m .. Ls + 2m+1]
    v2f ov; ov.x = s00 * d0 + s01 * d1; ov.y = s10 * d0 + s11 * d1;
    __builtin_nontemporal_store(ov, &out2[idx * (L_ / 2) + (LS_ / 2) + m]);
}

// -------------------------------------------------------------------------
// Kernel 2: coupled operator  scm[b] = W(1024x1024) * sc[b](1024x4096)
// using V_WMMA_F32_16X16X4_F32.  Block tile 128x128, 8 waves (2x4 grid),
// each wave owns a 64x32 sub-tile = 4x2 WMMA tiles of v8f accumulators.
// Both A and B tiles are stored K-minor in LDS (pad 20) so every fragment
// is a single aligned 8-byte DS load matching the wave32 f32 WMMA layout
// (lanes 0-15 hold K0/K1, lanes 16-31 hold K2/K3).
// -------------------------------------------------------------------------
__global__ __launch_bounds__(256)
void gemm_wmma_kernel(const float* __restrict__ W,    // [PC][PC]
                      const float* __restrict__ sc,   // [B][PC][LS]
                      float* __restrict__ scm) {      // [B][PC][LS]
    __shared__ float ldsA[BM * LDA];   // A tile, row-major [m][k], k-minor
    __shared__ float ldsB[BN * LDB];   // B tile, column-major [n][k], k-minor

    const int b    = blockIdx.z;
    const int row0 = blockIdx.y * BM;
    const int col0 = blockIdx.x * BN;
    const float* __restrict__ Xb = sc  + (size_t)b * PC_ * LS_;
    float* __restrict__       Yb = scm + (size_t)b * PC_ * LS_;

    const int tid  = threadIdx.x;
    const int wave = tid >> 5;
    const int lane = tid & 31;
    const int wm   = wave >> 2;      // 0..1  : 64-row slab
    const int wn   = wave & 3;       // 0..3  : 32-col slab
    const int tn   = lane & 15;      // N for B/C frags, M for A frag
    const int kh   = lane >> 4;      // which K-pair this half-wave holds

    v8f zero = {};
    v8f acc[4][2];
#pragma unroll
    for (int mi = 0; mi < 4; ++mi)
#pragma unroll
        for (int ni = 0; ni < 2; ++ni) acc[mi][ni] = zero;

    // staging indices (stage-invariant)
    const int ar = tid >> 1;             // A row 0..127 (2 threads/row)
    const int ak = (tid & 1) * 8;        // A col group {0,8}
    const int bc = tid & 127;            // B column 0..127
    const int bk0 = (tid >> 7) * 8;      // B k group {0,8}

    for (int k0 = 0; k0 < PC_; k0 += BK) {
        __syncthreads();
        // ---- Stage A (128x16, row-major) : two v4f per thread ----
        *(v4f*)&ldsA[ar * LDA + ak] =
            *(const v4f*)&W[(size_t)(row0 + ar) * PC_ + k0 + ak];
        *(v4f*)&ldsA[ar * LDA + ak + 4] =
            *(const v4f*)&W[(size_t)(row0 + ar) * PC_ + k0 + ak + 4];
        // ---- Stage B (16x128) transposed to column-major [n][k] ----
        // 4 coalesced scalar loads -> one aligned b128 DS store, twice.
#pragma unroll
        for (int g = 0; g < 2; ++g) {
            const int bk = bk0 + g * 4;
            v4f bv;
            bv.x = Xb[(size_t)(k0 + bk + 0) * LS_ + col0 + bc];
            bv.y = Xb[(size_t)(k0 + bk + 1) * LS_ + col0 + bc];
            bv.z = Xb[(size_t)(k0 + bk + 2) * LS_ + col0 + bc];
            bv.w = Xb[(size_t)(k0 + bk + 3) * LS_ + col0 + bc];
            *(v4f*)&ldsB[bc * LDB + bk] = bv;
        }
        // prefetch next stage's tiles into cache while we compute
        if (k0 + BK < PC_) {
            __builtin_prefetch(&W[(size_t)(row0 + ar) * PC_ + k0 + BK], 0, 3);
            __builtin_prefetch(&Xb[(size_t)(k0 + BK + (tid & 15)) * LS_ +
                                   col0 + (tid >> 4) * 8], 0, 3);
        }
        __syncthreads();

#pragma unroll
        for (int ks = 0; ks < BK; ks += 4) {
            const int kb = ks + kh * 2;        // K pair owned by this half-wave
            v2f bfrag[2];
#pragma unroll
            for (int ni = 0; ni < 2; ++ni) {
                const int cc = wn * 32 + ni * 16 + tn;
                bfrag[ni] = *(const v2f*)&ldsB[cc * LDB + kb];
            }
            v2f afrag[4];
#pragma unroll
            for (int mi = 0; mi < 4; ++mi) {
                const int rr = wm * 64 + mi * 16 + tn;
                afrag[mi] = *(const v2f*)&ldsA[rr * LDA + kb];
            }
#pragma unroll
            for (int mi = 0; mi < 4; ++mi)
#pragma unroll
                for (int ni = 0; ni < 2; ++ni)
                    acc[mi][ni] = __builtin_amdgcn_wmma_f32_16x16x4_f32(
                        false, afrag[mi], false, bfrag[ni],
                        (short)0, acc[mi][ni], false, false);
        }
    }

    // Store: lanes 0-15 hold M = r, lanes 16-31 hold M = r + 8 (N = lane&15).
#pragma unroll
    for (int mi = 0; mi < 4; ++mi) {
        const int rbase = row0 + wm * 64 + mi * 16 + (kh ? 8 : 0);
#pragma unroll
        for (int ni = 0; ni < 2; ++ni) {
            const int cc = col0 + wn * 32 + ni * 16 + tn;
#pragma unroll
            for (int r = 0; r < 8; ++r)
                Yb[(size_t)(rbase + r) * LS_ + cc] = acc[mi][ni][r];
        }
    }
}

// -------------------------------------------------------------------------
// Kernel 3: BatchNorm batch statistics (biased, over B x Ls per channel).
// One block per channel i; emits mean and rsqrt(var + eps).
// -------------------------------------------------------------------------
__global__ __launch_bounds__(256)
void bnstats_kernel(const float* __restrict__ scm, float* __restrict__ stats) {
    __shared__ float ssum[256];
    __shared__ float ssq[256];
    const int i = blockIdx.x;              // channel 0..1023
    float s = 0.f, q = 0.f;
    for (int b = 0; b < B_; ++b) {
        const float* row = scm + ((size_t)b * PC_ + i) * LS_;
        for (int l = threadIdx.x; l < LS_; l += 256) {
            const float v = row[l];
            s += v; q += v * v;
        }
    }
    ssum[threadIdx.x] = s;
    ssq[threadIdx.x]  = q;
    __syncthreads();
    for (int off = 128; off > 0; off >>= 1) {
        if (threadIdx.x < off) {
            ssum[threadIdx.x] += ssum[threadIdx.x + off];
            ssq[threadIdx.x]  += ssq[threadIdx.x + off];
        }
        __syncthreads();
    }
    if (threadIdx.x == 0) {
        const float n    = (float)(B_ * LS_);
        const float mean = ssum[0] / n;
        const float var  = ssq[0] / n - mean * mean;
        stats[i]        = mean;
        stats[PC_ + i]  = rsqrtf(var + 1e-5f);
    }
}

// -------------------------------------------------------------------------
// Kernel 4: BN + exact GELU + synthesis on adjacent scm pairs -> first half
// of the output (non-temporal store; out is never re-read).
// -------------------------------------------------------------------------
__global__ __launch_bounds__(256)
void bngelu_synth_kernel(const v2f* __restrict__ scm2,
                         const float* __restrict__ stats,
                         const float* __restrict__ gamma,
                         const float* __restrict__ beta,
                         const float* __restrict__ S,
                         v2f* __restrict__ out2) {
    const long t  = (long)blockIdx.x * blockDim.x + threadIdx.x; // B*PC*2048
    const int  n2 = (int)(t & 2047);
    const long r  = t >> 11;                // b*PC + i
    const int  i  = (int)(r & (PC_ - 1));
    const int  b  = (int)(r >> 10);
    const int  p  = i >> 7;                 // i / C
    const int  c  = i & (C_ - 1);

    const v2f v      = scm2[t];
    const float mean = stats[i];
    const float rinv = stats[PC_ + i];
    const float g    = gamma[i];
    const float be   = beta[i];

    float y0 = (v.x - mean) * rinv * g + be;
    float y1 = (v.y - mean) * rinv * g + be;
    // exact GELU
    y0 = 0.5f * y0 * (1.0f + erff(y0 * 0.70710678118654752f));
    y1 = 0.5f * y1 * (1.0f + erff(y1 * 0.70710678118654752f));

    const float s00 = S[0], s01 = S[1], s10 = S[2], s11 = S[3];
    v2f o;
    o.x = s00 * y0 + s01 * y1;
    o.y = s10 * y0 + s11 * y1;
    __builtin_nontemporal_store(
        o, &out2[((size_t)(p * B_ + b) * C_ + c) * (L_ / 2) + n2]);
}

// -------------------------------------------------------------------------
extern "C" void kernel_launch(void* const* d_in, const int* in_sizes, int n_in,
                              void* d_out, int out_size, void* d_ws, size_t ws_size,
                              hipStream_t stream) {
    const float* xs    = (const float*)d_in[0];
    const float* A     = (const float*)d_in[1];
    const float* S     = (const float*)d_in[2];
    const float* W     = (const float*)d_in[3];
    const float* gamma = (const float*)d_in[4];
    const float* beta  = (const float*)d_in[5];
    float* out = (float*)d_out;

    float* sc    = (float*)d_ws;                       // [B][PC][LS]
    float* scm   = sc  + (size_t)B_ * PC_ * LS_;       // [B][PC][LS]
    float* stats = scm + (size_t)B_ * PC_ * LS_;       // [2][PC]

    // 1) analysis + detail-half synthesis
    {
        const long n = (long)P_ * B_ * C_ * (L_ / 4);  // one v4f per thread
        analysis_kernel<<<(int)(n / 256), 256, 0, stream>>>(
            (const v4f*)xs, A, S, (v2f*)sc, (v2f*)out);
    }
    // 2) coupled GEMM with f32 WMMA
    {
        dim3 grid(LS_ / BN, PC_ / BM, B_);
        gemm_wmma_kernel<<<grid, 256, 0, stream>>>(W, sc, scm);
    }
    // 3) BN statistics
    bnstats_kernel<<<PC_, 256, 0, stream>>>(scm, stats);
    // 4) BN + GELU + scaling-half synthesis
    {
        const long n = (long)B_ * PC_ * (LS_ / 2);     // one v2f per thread
        bngelu_synth_kernel<<<(int)(n / 256), 256, 0, stream>>>(
            (const v2f*)scm, stats, gamma, beta, S, (v2f*)out);
    }
}